// MistralSelfAttention_8727373545684
// MI455X (gfx1250) — compile-verified
//
#include <hip/hip_runtime.h>
#include <hip/hip_bf16.h>
#include <math.h>

// Problem constants (match reference)
#define B_ 2
#define T_ 2048
#define H_ 32
#define D_ 128
#define QG_ 4
#define HKV_ 8
#define INNER_ 4096
#define KVI_ 1024
#define ATTN_SCALE_ 0.08838834764831845f   // 1/sqrt(128)

typedef float v2f __attribute__((ext_vector_type(2)));
typedef float v8f __attribute__((ext_vector_type(8)));

// ---------------------------------------------------------------------------
// Async global->LDS staging (CDNA5 GLOBAL_LOAD_ASYNC_TO_LDS_B128, ASYNCcnt).
// Builtin signature on this toolchain (from clang diagnostic):
//   (int4-vector addrspace(1)* src, int4-vector addrspace(3)* dst, Ii, Ii)
// ---------------------------------------------------------------------------
#if defined(__has_builtin)
#if __has_builtin(__builtin_amdgcn_global_load_async_to_lds_b128)
#define GEMM_ASYNC 1
#endif
#endif

typedef int v4i_vs __attribute__((vector_size(4 * sizeof(int))));
typedef __attribute__((address_space(1))) v4i_vs* gbl_v4i_ptr;
typedef __attribute__((address_space(3))) v4i_vs* lds_v4i_ptr;

__device__ __forceinline__ lds_v4i_ptr to_lds(void* p) {
    // flat->LDS: low 32 bits of the flat address are the LDS byte address
    return (lds_v4i_ptr)(unsigned int)(unsigned long long)p;
}
__device__ __forceinline__ gbl_v4i_ptr to_gbl(const void* p) {
    return (gbl_v4i_ptr)(unsigned long long)p;
}

#if defined(__has_builtin) && __has_builtin(__builtin_amdgcn_s_wait_asynccnt)
#define WAIT_ASYNC(n) __builtin_amdgcn_s_wait_asynccnt(n)
#else
#define WAIT_ASYNC(n) asm volatile("s_wait_asynccnt %0" :: "i"(n) : "memory")
#endif

// ---------------------------------------------------------------------------
// fp32 GEMM  C[M,N] = A[M,K] @ W[K,N]  via V_WMMA_F32_16X16X4_F32.
// Block = 256 threads (8 waves, 2x4). Block tile 64(M) x 128(N); wave tile
// 32x32 (4 v8f accumulators) => 1.5 LDS loads per WMMA. K staged in chunks
// of 32 through double-buffered LDS filled by async b128 loads.
// LDS bank padding:
//   A stride 36 floats: bank = 36*row + k (distinct across 16 rows, even-based
//     so the hi-half's +2 column offset cannot collide).
//   B stride 136 floats: hi-half reads +2 rows => +16 banks, disjoint from the
//     16-consecutive-column window of the low half.
// ---------------------------------------------------------------------------
#define AST 36
#define BST 136
#define KCH 32

__device__ __forceinline__ void stage_chunk(
    const float* __restrict__ A, const float* __restrict__ W,
    float* lAbuf, float* lBbuf, int tid, int m0, int n0, int k0, int K, int N)
{
#ifdef GEMM_ASYNC
    // A chunk: 64 x 32 floats = 512 float4 => 2 per thread
    #pragma unroll
    for (int i = 0; i < 2; ++i) {
        const int e = (tid + i * 256) * 4;
        const int r = e >> 5, c = e & 31;
        const float* g = A + (size_t)(m0 + r) * K + k0 + c;
        __builtin_amdgcn_global_load_async_to_lds_b128(
            to_gbl(g), to_lds(lAbuf + r * AST + c), 0, 0);
    }
    // B chunk: 32 x 128 floats = 1024 float4 => 4 per thread
    #pragma unroll
    for (int i = 0; i < 4; ++i) {
        const int e = (tid + i * 256) * 4;
        const int r = e >> 7, c = e & 127;
        const float* g = W + (size_t)(k0 + r) * N + n0 + c;
        __builtin_amdgcn_global_load_async_to_lds_b128(
            to_gbl(g), to_lds(lBbuf + r * BST + c), 0, 0);
    }
#else
    #pragma unroll
    for (int i = 0; i < 2; ++i) {
        const int e = (tid + i * 256) * 4;
        const int r = e >> 5, c = e & 31;
        *(float4*)(lAbuf + r * AST + c) =
            *(const float4*)(A + (size_t)(m0 + r) * K + k0 + c);
    }
    #pragma unroll
    for (int i = 0; i < 4; ++i) {
        const int e = (tid + i * 256) * 4;
        const int r = e >> 7, c = e & 127;
        *(float4*)(lBbuf + r * BST + c) =
            *(const float4*)(W + (size_t)(k0 + r) * N + n0 + c);
    }
#endif
}

__global__ __launch_bounds__(256) void gemm_f32_wmma(
    const float* __restrict__ A, const float* __restrict__ W,
    float* __restrict__ C, int M, int N, int K)
{
    __shared__ float lA[2][64 * AST];
    __shared__ float lB[2][KCH * BST];

    const int tid  = threadIdx.x;
    const int wave = tid >> 5;
    const int lane = tid & 31;
    const int lm   = lane & 15;   // A: row m; B/C/D: col n (CDNA5 WMMA layout)
    const int hi   = lane >> 4;   // K sub-offset selector (2*hi)
    const int wm   = wave >> 2;   // 0..1  (M direction)
    const int wn   = wave & 3;    // 0..3  (N direction)

    const int m0 = blockIdx.y * 64;
    const int n0 = blockIdx.x * 128;

    v8f acc[2][2] = {};   // [m-subtile][n-subtile]

    const int nk = K / KCH;

#ifdef GEMM_ASYNC
    stage_chunk(A, W, lA[0], lB[0], tid, m0, n0, 0, K, N);
#endif

    for (int kc = 0; kc < nk; ++kc) {
        const int cur = kc & 1;
#ifdef GEMM_ASYNC
        if (kc + 1 < nk) {
            stage_chunk(A, W, lA[cur ^ 1], lB[cur ^ 1], tid, m0, n0,
                        (kc + 1) * KCH, K, N);
            WAIT_ASYNC(6);   // 6 older ops = current chunk retired (in-order)
        } else {
            WAIT_ASYNC(0);
        }
#else
        stage_chunk(A, W, lA[cur], lB[cur], tid, m0, n0, kc * KCH, K, N);
#endif
        __syncthreads();

        const float* pA = lA[cur];
        const float* pB = lB[cur];
        #pragma unroll
        for (int kk = 0; kk < KCH / 4; ++kk) {
            const int ka = kk * 4 + 2 * hi;
            v2f a0 = *(const v2f*)(pA + (wm * 32 + lm) * AST + ka);
            v2f a1 = *(const v2f*)(pA + (wm * 32 + 16 + lm) * AST + ka);
            #pragma unroll
            for (int j = 0; j < 2; ++j) {
                const int col = wn * 32 + j * 16 + lm;
                v2f b;
                b.x = pB[(ka + 0) * BST + col];
                b.y = pB[(ka + 1) * BST + col];
                acc[0][j] = __builtin_amdgcn_wmma_f32_16x16x4_f32(
                    false, a0, false, b, (short)0, acc[0][j], false, false);
                acc[1][j] = __builtin_amdgcn_wmma_f32_16x16x4_f32(
                    false, a1, false, b, (short)0, acc[1][j], false, false);
            }
        }
        __syncthreads();   // all waves done reading before buffer is refilled
    }

    // C/D layout: vgpr r, lane => element [r + 8*hi][lm]
    #pragma unroll
    for (int mt = 0; mt < 2; ++mt)
        #pragma unroll
        for (int j = 0; j < 2; ++j)
            #pragma unroll
            for (int r = 0; r < 8; ++r) {
                const int row = m0 + wm * 32 + mt * 16 + r + 8 * hi;
                C[(size_t)row * N + n0 + wn * 32 + j * 16 + lm] = acc[mt][j][r];
            }
}

// ---------------------------------------------------------------------------
// RoPE (in-place), layout (B, T, Hx, D). One thread per rotation pair.
// ---------------------------------------------------------------------------
__global__ __launch_bounds__(256) void rope_kernel(
    float* __restrict__ x, const float* __restrict__ cosb,
    const float* __restrict__ sinb, int Hx, int total)
{
    const int idx = blockIdx.x * blockDim.x + threadIdx.x;
    if (idx >= total) return;
    const int d = idx & 63;
    const int h = (idx >> 6) % Hx;
    const int t = (idx / (64 * Hx)) % T_;
    const int b = idx / (64 * Hx * T_);
    const size_t base = (((size_t)b * T_ + t) * Hx + h) * D_;
    const float c0 = cosb[t * D_ + d],      s0 = sinb[t * D_ + d];
    const float c1 = cosb[t * D_ + d + 64], s1 = sinb[t * D_ + d + 64];
    const float lo = x[base + d];
    const float hv = x[base + d + 64];
    x[base + d]      = lo * c0 - hv * s0;
    x[base + d + 64] = hv * c1 + lo * s1;
}

// ---------------------------------------------------------------------------
// Causal flash attention with GQA. One wave owns one (b, h, 16-row) tile of
// the 16x128 output. S = Q@K^T via WMMA, online softmax with 16-lane shuffle
// reductions (row groups coincide with the C/D VGPR layout, so accumulator
// rescale is lane-local), P re-laid-out C->A through LDS, O += P@V via WMMA.
// ---------------------------------------------------------------------------
__global__ __launch_bounds__(256) void attn_kernel(
    const float* __restrict__ Q, const float* __restrict__ Kt,
    const float* __restrict__ Vt, float* __restrict__ O)
{
    __shared__ float lp[8][16 * 17];   // per-wave P tile, padded stride 17

    const int tid  = threadIdx.x;
    const int wave = tid >> 5;
    const int lane = tid & 31;
    const int lm   = lane & 15;
    const int hi   = lane >> 4;

    const int mb  = blockIdx.x * 8 + wave;   // 0..T/16-1
    const int h   = blockIdx.y;
    const int b   = blockIdx.z;
    const int hkv = h / QG_;
    const int mq0 = mb * 16;

    // Preload Q tile in A-layout: lane holds Q[lm][4s + 2hi + {0,1}], s=0..31.
    v2f qreg[32];
    {
        const float* qrow = Q + (((size_t)b * T_ + mq0 + lm) * H_ + h) * D_ + 2 * hi;
        #pragma unroll
        for (int s = 0; s < 32; ++s)
            qreg[s] = *(const v2f*)(qrow + 4 * s);
    }

    v8f oacc[8] = {};
    float mx[8], lsum[8];
    #pragma unroll
    for (int r = 0; r < 8; ++r) { mx[r] = -3.0e38f; lsum[r] = 0.0f; }

    float* P = &lp[wave][0];
    const size_t kvstride = (size_t)HKV_ * D_;

    for (int kb = 0; kb <= mb; ++kb) {
        const int kt0 = kb * 16;

        // ---- S = Q @ K^T : B[kk][n] = K[kt0+n][kk], contiguous along D.
        v8f s = {};
        {
            const float* krow = Kt + (((size_t)b * T_ + kt0 + lm) * HKV_ + hkv) * D_ + 2 * hi;
            #pragma unroll
            for (int st = 0; st < 32; ++st) {
                v2f bv = *(const v2f*)(krow + 4 * st);
                s = __builtin_amdgcn_wmma_f32_16x16x4_f32(
                    false, qreg[st], false, bv, (short)0, s, false, false);
            }
        }

        // ---- scale + causal mask + online softmax (rows = r + 8*hi)
        float alpha[8];
        #pragma unroll
        for (int r = 0; r < 8; ++r) {
            float v = s[r] * ATTN_SCALE_;
            const int row = mq0 + r + 8 * hi;
            const int col = kt0 + lm;
            if (col > row) v = -3.0e38f;           // exp -> 0 == -30000 mask

            float rm = v;
            #pragma unroll
            for (int off = 1; off < 16; off <<= 1)
                rm = fmaxf(rm, __shfl_xor(rm, off, 32));
            const float mnew = fmaxf(mx[r], rm);
            const float a = __expf(mx[r] - mnew);
            const float p = __expf(v - mnew);
            float rs = p;
            #pragma unroll
            for (int off = 1; off < 16; off <<= 1)
                rs += __shfl_xor(rs, off, 32);
            lsum[r] = lsum[r] * a + rs;
            mx[r]   = mnew;
            alpha[r] = a;
            P[(r + 8 * hi) * 17 + lm] = p;         // store P in C-layout
        }
        asm volatile("s_wait_dscnt 0" ::: "memory");

        // ---- rescale accumulators (lane-local: O rows == S rows per lane)
        #pragma unroll
        for (int nt = 0; nt < 8; ++nt)
            #pragma unroll
            for (int r = 0; r < 8; ++r)
                oacc[nt][r] *= alpha[r];

        // ---- reload P in A-layout
        v2f pa[4];
        #pragma unroll
        for (int st = 0; st < 4; ++st) {
            pa[st].x = P[lm * 17 + 4 * st + 2 * hi];
            pa[st].y = P[lm * 17 + 4 * st + 2 * hi + 1];
        }

        // ---- O += P @ V : B[kk][n] = V[kt0+kk][d0+n]
        const float* vbase = Vt + (((size_t)b * T_ + kt0) * HKV_ + hkv) * D_;
        #pragma unroll
        for (int nt = 0; nt < 8; ++nt) {
            v8f acc = oacc[nt];
            #pragma unroll
            for (int st = 0; st < 4; ++st) {
                const int kk = 4 * st + 2 * hi;
                v2f bv;
                bv.x = vbase[(size_t)(kk    ) * kvstride + nt * 16 + lm];
                bv.y = vbase[(size_t)(kk + 1) * kvstride + nt * 16 + lm];
                acc = __builtin_amdgcn_wmma_f32_16x16x4_f32(
                    false, pa[st], false, bv, (short)0, acc, false, false);
            }
            oacc[nt] = acc;
        }
        asm volatile("s_wait_dscnt 0" ::: "memory");
    }

    // ---- normalize and write O (B, T, H, D)
    #pragma unroll
    for (int nt = 0; nt < 8; ++nt) {
        #pragma unroll
        for (int r = 0; r < 8; ++r) {
            const int row = mq0 + r + 8 * hi;
            O[(((size_t)b * T_ + row) * H_ + h) * D_ + nt * 16 + lm] =
                oacc[nt][r] / lsum[r];
        }
    }
}

// ---------------------------------------------------------------------------
// Host-side launch. Inputs (all fp32, reference order):
//   0: stm (B,T,H,D)  1: wq (4096,4096)  2: wk (4096,1024)  3: wv (4096,1024)
//   4: wo (4096,4096) 5: cos (T,D)       6: sin (T,D)       7: mask_w  8: mask_b
// Workspace: Q (64MB) | K (16MB) | V (16MB) | O (64MB) = 160MB fp32.
// ---------------------------------------------------------------------------
extern "C" void kernel_launch(void* const* d_in, const int* in_sizes, int n_in,
                              void* d_out, int out_size, void* d_ws, size_t ws_size,
                              hipStream_t stream) {
    (void)in_sizes; (void)n_in; (void)out_size; (void)ws_size;

    const float* stm  = (const float*)d_in[0];
    const float* wq   = (const float*)d_in[1];
    const float* wk   = (const float*)d_in[2];
    const float* wv   = (const float*)d_in[3];
    const float* wo   = (const float*)d_in[4];
    const float* cosb = (const float*)d_in[5];
    const float* sinb = (const float*)d_in[6];
    float* out = (float*)d_out;

    float* ws = (float*)d_ws;
    float* Qb = ws;                                     // 4096 x 4096
    float* Kb = Qb + (size_t)B_ * T_ * H_ * D_;         // 4096 x 1024
    float* Vb = Kb + (size_t)B_ * T_ * HKV_ * D_;       // 4096 x 1024
    float* Ob = Vb + (size_t)B_ * T_ * HKV_ * D_;       // 4096 x 4096

    const int M = B_ * T_;                              // 4096

    // QKV projections
    gemm_f32_wmma<<<dim3(INNER_ / 128, M / 64), 256, 0, stream>>>(stm, wq, Qb, M, INNER_, INNER_);
    gemm_f32_wmma<<<dim3(KVI_   / 128, M / 64), 256, 0, stream>>>(stm, wk, Kb, M, KVI_,   INNER_);
    gemm_f32_wmma<<<dim3(KVI_   / 128, M / 64), 256, 0, stream>>>(stm, wv, Vb, M, KVI_,   INNER_);

    // RoPE on Q and K
    {
        const int tq = B_ * T_ * H_   * (D_ / 2);   // 8,388,608
        const int tk = B_ * T_ * HKV_ * (D_ / 2);   // 2,097,152
        rope_kernel<<<(tq + 255) / 256, 256, 0, stream>>>(Qb, cosb, sinb, H_,   tq);
        rope_kernel<<<(tk + 255) / 256, 256, 0, stream>>>(Kb, cosb, sinb, HKV_, tk);
    }

    // Causal GQA flash attention
    attn_kernel<<<dim3(T_ / 16 / 8, H_, B_), 256, 0, stream>>>(Qb, Kb, Vb, Ob);

    // Output projection
    gemm_f32_wmma<<<dim3(INNER_ / 128, M / 64), 256, 0, stream>>>(Ob, wo, out, M, INNER_, INNER_);
}